// ModelNew_1580547974612
// MI455X (gfx1250) — compile-verified
//
#include <hip/hip_runtime.h>
#include <hip/hip_bf16.h>

typedef float v2f __attribute__((ext_vector_type(2)));
typedef float v8f __attribute__((ext_vector_type(8)));

#define N_BATCH 32
#define CIN     64
#define COUT    128
#define HW      4096                    // 64*64
#define INV_AREA (1.0f / (130.0f * 130.0f))
#define EPS 1e-5f

// ---------------------------------------------------------------------------
// Kernel 1: S_x[n*64+cin] = sum over 64x64 spatial plane of x.
// One 256-thread block per plane; each thread reads 4 float4 (coalesced 4KB
// sweeps), LDS tree-reduce. 33.5MB total read -> HBM-bound, ~1.4us.
// ---------------------------------------------------------------------------
__global__ void reduce_x_kernel(const float* __restrict__ x,
                                float* __restrict__ S_x) {
    const int plane = blockIdx.x;            // 0..2047  (n*64 + cin)
    const int t = threadIdx.x;               // 0..255
    const float4* xv = reinterpret_cast<const float4*>(x) + (size_t)plane * (HW / 4);

    float s = 0.0f;
#pragma unroll
    for (int j = 0; j < 4; ++j) {
        float4 v = xv[t + j * 256];
        s += v.x + v.y + v.z + v.w;
    }

    __shared__ float red[256];
    red[t] = s;
    __syncthreads();
#pragma unroll
    for (int off = 128; off > 0; off >>= 1) {
        if (t < off) red[t] += red[t + off];
        __syncthreads();
    }
    if (t == 0) S_x[plane] = red[0];
}

// ---------------------------------------------------------------------------
// Kernel 2: S_w[cin*128+cout] = sum of the 16 kernel taps (contiguous).
// One thread per (cin,cout); 16 contiguous floats = 4 float4 loads.
// ---------------------------------------------------------------------------
__global__ void reduce_w_kernel(const float* __restrict__ w,
                                float* __restrict__ S_w) {
    const int idx = blockIdx.x * blockDim.x + threadIdx.x;   // 0..8191
    const float4* wv = reinterpret_cast<const float4*>(w) + (size_t)idx * 4;
    float s = 0.0f;
#pragma unroll
    for (int j = 0; j < 4; ++j) {
        float4 v = wv[j];
        s += v.x + v.y + v.z + v.w;
    }
    S_w[idx] = s;
}

// ---------------------------------------------------------------------------
// Kernel 3: M = S_x(32x64) @ S_w(64x128) via V_WMMA_F32_16X16X4_F32,
// then m = M*INV_AREA + bias, layernorm -> hardtanh -> layernorm.
// One block, 512 threads = 16 wave32s; wave w owns output tile
// (m_tile = w>>3, n_tile = w&7), accumulating 16 k-steps of K=4.
// ---------------------------------------------------------------------------
__global__ void
__launch_bounds__(512)
gemm_ln_kernel(const float* __restrict__ S_x,
               const float* __restrict__ S_w,
               const float* __restrict__ bias,
               const float* __restrict__ gamma1, const float* __restrict__ beta1,
               const float* __restrict__ gamma2, const float* __restrict__ beta2,
               float* __restrict__ out) {
    __shared__ float m_sh[N_BATCH * COUT];   // 16 KB
    __shared__ float h_sh[N_BATCH * COUT];   // 16 KB

    const int tid  = threadIdx.x;
    const int wid  = tid >> 5;               // 0..15
    const int lane = tid & 31;
    const int m_tile = wid >> 3;             // 0..1  (rows 0-15 / 16-31)
    const int n_tile = wid & 7;              // 0..7  (cols of 16)

    const int lmod = lane & 15;
    const int lhi  = lane >> 4;              // 0 or 1

    v8f acc = {};
#pragma unroll
    for (int k0 = 0; k0 < CIN; k0 += 4) {
        const int kb = k0 + 2 * lhi;         // K index of element .x
        // A fragment: 16x4 f32, lane holds M = lmod, K = kb, kb+1
        v2f a;
        a.x = S_x[(m_tile * 16 + lmod) * CIN + kb];
        a.y = S_x[(m_tile * 16 + lmod) * CIN + kb + 1];
        // B fragment: 4x16 f32, lane holds N = lmod, K = kb, kb+1
        v2f b;
        b.x = S_w[kb * COUT + n_tile * 16 + lmod];
        b.y = S_w[(kb + 1) * COUT + n_tile * 16 + lmod];
        acc = __builtin_amdgcn_wmma_f32_16x16x4_f32(
            /*neg_a=*/false, a, /*neg_b=*/false, b,
            /*c_mod=*/(short)0, acc, /*reuse_a=*/false, /*reuse_b=*/false);
    }

    // Scatter accumulator: VGPR i -> M = i + 8*lhi, N = lmod (within tile)
    const int c = n_tile * 16 + lmod;
    const float bc = bias[c];
#pragma unroll
    for (int i = 0; i < 8; ++i) {
        const int row = m_tile * 16 + 8 * lhi + i;
        m_sh[row * COUT + c] = acc[i] * INV_AREA + bc;
    }
    __syncthreads();

    // Epilogue: per-row (128-ch) layernorm -> clip -> layernorm. 32 lanes.
    if (tid < N_BATCH) {
        const int row = tid;
        const float* mr = &m_sh[row * COUT];

        float mu = 0.0f;
        for (int i = 0; i < COUT; ++i) mu += mr[i];
        mu *= (1.0f / COUT);
        float var = 0.0f;
        for (int i = 0; i < COUT; ++i) { float d = mr[i] - mu; var += d * d; }
        var *= (1.0f / COUT);
        const float r1 = rsqrtf(var + EPS);

        float* hr = &h_sh[row * COUT];
        float mu2 = 0.0f;
        for (int i = 0; i < COUT; ++i) {
            float h = (mr[i] - mu) * r1 * gamma1[i] + beta1[i];
            h = fminf(1.0f, fmaxf(-1.0f, h));
            hr[i] = h;
            mu2 += h;
        }
        mu2 *= (1.0f / COUT);
        float var2 = 0.0f;
        for (int i = 0; i < COUT; ++i) { float d = hr[i] - mu2; var2 += d * d; }
        var2 *= (1.0f / COUT);
        const float r2 = rsqrtf(var2 + EPS);

        for (int i = 0; i < COUT; ++i)
            out[row * COUT + i] = (hr[i] - mu2) * r2 * gamma2[i] + beta2[i];
    }
}

// ---------------------------------------------------------------------------
extern "C" void kernel_launch(void* const* d_in, const int* in_sizes, int n_in,
                              void* d_out, int out_size, void* d_ws, size_t ws_size,
                              hipStream_t stream) {
    const float* x      = (const float*)d_in[0];   // 32*64*64*64
    const float* weight = (const float*)d_in[1];   // 64*128*4*4
    const float* bias   = (const float*)d_in[2];   // 128
    const float* gamma1 = (const float*)d_in[3];
    const float* beta1  = (const float*)d_in[4];
    const float* gamma2 = (const float*)d_in[5];
    const float* beta2  = (const float*)d_in[6];
    float* out = (float*)d_out;                    // 32*128

    float* S_x = (float*)d_ws;                     // 2048 floats
    float* S_w = S_x + N_BATCH * CIN;              // 8192 floats

    reduce_x_kernel<<<N_BATCH * CIN, 256, 0, stream>>>(x, S_x);
    reduce_w_kernel<<<(CIN * COUT) / 256, 256, 0, stream>>>(weight, S_w);
    gemm_ln_kernel<<<1, 512, 0, stream>>>(S_x, S_w, bias,
                                          gamma1, beta1, gamma2, beta2, out);
}